// TSVDModule_34789235097732
// MI455X (gfx1250) — compile-verified
//
#include <hip/hip_runtime.h>

typedef float v2f __attribute__((ext_vector_type(2)));
typedef float v8f __attribute__((ext_vector_type(8)));
typedef unsigned int u32x4 __attribute__((ext_vector_type(4)));
typedef int i32x4 __attribute__((ext_vector_type(4)));
typedef int i32x8 __attribute__((ext_vector_type(8)));
typedef __attribute__((address_space(3))) float lds_f32;

#if __has_builtin(__builtin_amdgcn_tensor_load_to_lds) && \
    __has_builtin(__builtin_amdgcn_s_wait_tensorcnt)
#define TSVD_USE_TDM 1
#else
#define TSVD_USE_TDM 0
#endif

namespace {
constexpr int kB   = 32;           // batch (columns of A)
constexpr int kC   = 256;          // channels (rows of A)
constexpr int kH   = 56;
constexpr int kW   = 56;
constexpr int kM   = kH * kW;      // 3136 independent matrices
constexpr int kQ   = 8;            // low rank
constexpr int kMat = kC * kB;      // 8192 floats per matrix (32 KB)
}

// ---------------------------------------------------------------------------
// 8x8 Cholesky (upper R, G = R^T R), run by a single lane on LDS data.
// ---------------------------------------------------------------------------
__device__ inline void cholesky8(float (*G)[8], float (*R)[8]) {
  for (int k = 0; k < 8; ++k) {
    float s = G[k][k];
    for (int i = 0; i < k; ++i) s -= R[i][k] * R[i][k];
    float d = sqrtf(fmaxf(s, 1e-20f));
    R[k][k] = d;
    float inv = 1.0f / d;
    for (int j = k + 1; j < 8; ++j) {
      float s2 = G[k][j];
      for (int i = 0; i < k; ++i) s2 -= R[i][k] * R[i][j];
      R[k][j] = s2 * inv;
    }
  }
}

// ---------------------------------------------------------------------------
// CholeskyQR of a 32x8 matrix sYv -> orthonormal columns into sQm[32][16]
// (cols 8..15 stay zero so WMMA B-operands see a zero-padded N=16).
// ---------------------------------------------------------------------------
__device__ inline void chol_orth32(int t, float (*sYv)[8], float (*sQm)[16],
                                   float (*sG)[8], float (*sR)[8]) {
  __syncthreads();
  if (t < 64) {
    int i = t >> 3, j = t & 7;
    float g = 0.f;
    for (int b = 0; b < kB; ++b) g = fmaf(sYv[b][i], sYv[b][j], g);
    sG[i][j] = g;
  }
  __syncthreads();
  if (t == 0) cholesky8(sG, sR);
  __syncthreads();
  if (t < kB) {
    float q[8];
#pragma unroll
    for (int j = 0; j < 8; ++j) {
      float v = sYv[t][j];
      for (int i = 0; i < j; ++i) v -= q[i] * sR[i][j];
      q[j] = v / sR[j][j];
      sQm[t][j] = q[j];
    }
  }
  __syncthreads();
}

// ---------------------------------------------------------------------------
// CholeskyQR of a 256x8 matrix sZ (rows padded to 9), normalized in place.
// Gram matrix built with a 4-way c-split across all 256 threads.
// ---------------------------------------------------------------------------
__device__ inline void chol_orth256(int t, float (*sZ)[9], float (*sG)[8],
                                    float (*sR)[8], float (*sGp)[64]) {
  __syncthreads();
  {
    int pr = t >> 2, part = t & 3;
    int i = pr >> 3, j = pr & 7;
    int c0 = part * 64;
    float g = 0.f;
    for (int c = c0; c < c0 + 64; ++c) g = fmaf(sZ[c][i], sZ[c][j], g);
    sGp[part][pr] = g;
  }
  __syncthreads();
  if (t < 64) sG[t >> 3][t & 7] = sGp[0][t] + sGp[1][t] + sGp[2][t] + sGp[3][t];
  __syncthreads();
  if (t == 0) cholesky8(sG, sR);
  __syncthreads();
  {
    float q[8];
#pragma unroll
    for (int j = 0; j < 8; ++j) {
      float v = sZ[t][j];
      for (int i = 0; i < j; ++i) v -= q[i] * sR[i][j];
      q[j] = v / sR[j][j];
    }
#pragma unroll
    for (int j = 0; j < 8; ++j) sZ[t][j] = q[j];
  }
  __syncthreads();
}

// ---------------------------------------------------------------------------
// Z[256x8] = A[256x32] * Qm[32x8(pad16)]  via v_wmma_f32_16x16x4_f32.
// 16 M-tiles over 8 waves, K=32 in 8 steps of 4. Uniform control flow so
// EXEC is all-ones at every WMMA.
// ---------------------------------------------------------------------------
__device__ inline void wmma_AxQ(int t, float (*sA)[33], float (*sQm)[16],
                                float (*sZ)[9]) {
  const int wv = t >> 5;        // wave id 0..7 (wave32)
  const int ln = t & 31;
  const int lm = ln & 15;       // M (or N) index within tile
  const int lh = ln >> 4;       // lane half selects K offset
#pragma unroll
  for (int tile = 0; tile < 2; ++tile) {
    const int c0 = (wv + tile * 8) * 16;
    v8f acc = {0.f, 0.f, 0.f, 0.f, 0.f, 0.f, 0.f, 0.f};
#pragma unroll
    for (int kk = 0; kk < 8; ++kk) {
      const int k0 = kk * 4 + 2 * lh;
      v2f a, b;
      a.x = sA[c0 + lm][k0];        // A frag: (M=lm, K=k0 + vgpr)
      a.y = sA[c0 + lm][k0 + 1];
      b.x = sQm[k0][lm];            // B frag: (K=k0 + vgpr, N=lm)
      b.y = sQm[k0 + 1][lm];
      acc = __builtin_amdgcn_wmma_f32_16x16x4_f32(
          false, a, false, b, (short)0, acc, false, false);
    }
    if (lm < kQ) {                  // cols 8..15 are zero padding
#pragma unroll
      for (int v = 0; v < 8; ++v) sZ[c0 + v + 8 * lh][lm] = acc[v];
    }
  }
}

// ---------------------------------------------------------------------------
// y[256x32] = P[256x8] * Qm^T[8x32], streamed straight to global memory
// (over A's slot — A only lives in LDS by now). 32 tiles over 8 waves.
// ---------------------------------------------------------------------------
__device__ inline void wmma_PQt_store(int t, float (*sP)[9], float (*sQm)[16],
                                      float* __restrict__ gOut) {
  const int wv = t >> 5;
  const int ln = t & 31;
  const int lm = ln & 15;
  const int lh = ln >> 4;
#pragma unroll
  for (int tile = 0; tile < 4; ++tile) {
    const int id = wv + tile * 8;       // 0..31
    const int c0 = (id >> 1) * 16;
    const int b0 = (id & 1) * 16;
    v8f acc = {0.f, 0.f, 0.f, 0.f, 0.f, 0.f, 0.f, 0.f};
#pragma unroll
    for (int kk = 0; kk < 2; ++kk) {    // K = 8
      const int k0 = kk * 4 + 2 * lh;
      v2f a, b;
      a.x = sP[c0 + lm][k0];            // P frag: (M=lm, K)
      a.y = sP[c0 + lm][k0 + 1];
      b.x = sQm[b0 + lm][k0];           // (Q^T)[K][N] = Q[b0+N][K]
      b.y = sQm[b0 + lm][k0 + 1];
      acc = __builtin_amdgcn_wmma_f32_16x16x4_f32(
          false, a, false, b, (short)0, acc, false, false);
    }
#pragma unroll
    for (int v = 0; v < 8; ++v)
      gOut[(c0 + v + 8 * lh) * kB + b0 + lm] = acc[v];  // 64B runs, coalesced
  }
}

// ---------------------------------------------------------------------------
// Kernel 1: x[B,C,H,W] -> A[m][c][b] (contiguous 32KB per matrix).
// One block per (c,h); reads coalesced along w, writes 128B runs along b.
// ---------------------------------------------------------------------------
__global__ __launch_bounds__(256) void tsvd_transpose_in(
    const float* __restrict__ x, float* __restrict__ A) {
  const int c = blockIdx.x / kH;
  const int h = blockIdx.x % kH;
  __shared__ float tile[kB][kW + 1];
  const int t = threadIdx.x;
  for (int i = t; i < kB * kW; i += 256) {
    int b = i / kW, w = i % kW;
    tile[b][w] = x[((size_t)b * kC + c) * (kH * kW) + h * kW + w];
  }
  __syncthreads();
  for (int i = t; i < kB * kW; i += 256) {
    int w = i >> 5, b = i & 31;
    A[(size_t)(h * kW + w) * kMat + c * kB + b] = tile[b][w];
  }
}

// ---------------------------------------------------------------------------
// Kernel 3: y[m][c][b] -> out[B,C,H,W].
// ---------------------------------------------------------------------------
__global__ __launch_bounds__(256) void tsvd_transpose_out(
    const float* __restrict__ Y, float* __restrict__ out) {
  const int c = blockIdx.x / kH;
  const int h = blockIdx.x % kH;
  __shared__ float tile[kB][kW + 1];
  const int t = threadIdx.x;
  for (int i = t; i < kB * kW; i += 256) {
    int w = i >> 5, b = i & 31;
    tile[b][w] = Y[(size_t)(h * kW + w) * kMat + c * kB + b];
  }
  __syncthreads();
  for (int i = t; i < kB * kW; i += 256) {
    int b = i / kW, w = i % kW;
    out[((size_t)b * kC + c) * (kH * kW) + h * kW + w] = tile[b][w];
  }
}

// ---------------------------------------------------------------------------
// Kernel 2: one block (8 wave32) per matrix. A is DMA'd into LDS by the
// Tensor Data Mover (pad feature reproduces the [256][33] anti-conflict
// layout in hardware); waves stage omega / zero Q padding concurrently.
// CholeskyQR replaces Householder QR (identical projector QQ^T).
// Reads A from AY and overwrites the same slot with y = A Q Q^T.
// ---------------------------------------------------------------------------
__global__ __launch_bounds__(256) void tsvd_block_kernel(
    float* __restrict__ AY, const float* __restrict__ omega) {
  __shared__ __align__(16) float sA[kC][33];  // A[c][b], padded rows
  __shared__ float sZ[kC][9];      // Z / P [c][q], padded
  __shared__ float sOm[kC][kQ];    // omega staged
  __shared__ float sQm[kB][16];    // Q (B-side basis), zero-padded to N=16
  __shared__ float sYv[kB][kQ];
  __shared__ float sG[8][8];
  __shared__ float sR[8][8];
  __shared__ float sGp[4][64];

  const int t = threadIdx.x;
  float* gA = AY + (size_t)blockIdx.x * kMat;

#if TSVD_USE_TDM
  // --- TDM: one descriptor DMAs the whole 32KB matrix global->LDS.
  // pad_enable with pad_interval=4 (every 32 DWORDs) and pad_amount=0
  // (1 DWORD) lands rows directly in the padded sA[256][33] layout.
  // TDM ignores EXEC; issue from wave 0 only (wave-uniform branch). ---
  if ((t >> 5) == 0) {
    unsigned lds_base =
        (unsigned)(unsigned long long)(lds_f32*)(&sA[0][0]);
    unsigned long long ga = (unsigned long long)gA;
    u32x4 g0 = {0u, 0u, 0u, 0u};
    g0[0] = 1u;                                        // count=1 (valid, user)
    g0[1] = lds_base;                                  // lds_addr (bytes)
    g0[2] = (unsigned)(ga & 0xFFFFFFFFull);            // global_addr[31:0]
    g0[3] = (unsigned)((ga >> 32) & 0x01FFFFFFull)     // global_addr[56:32]
          | (2u << 30);                                // type = 2 ("image")
    i32x8 g1 = {0, 0, 0, 0, 0, 0, 0, 0};
    g1[0] = (int)((2u << 16)        // data_size = 4 bytes
                | (1u << 20)        // pad_enable
                | (4u << 22));      // pad_interval: 32 DWORDs (pad_amount=0 -> 1 DWORD)
    g1[1] = (int)((unsigned)kB << 16);   // tensor_dim0 = 32 (b, contiguous)
    g1[2] = (int)((unsigned)kC << 16);   // tensor_dim1 = 256 (c)
    g1[3] = (int)((unsigned)kB << 16);   // tile_dim0   = 32
    g1[4] = kC;                          // tile_dim1   = 256
    g1[5] = kB;                          // tensor_dim0_stride = 32 elements
    i32x4 gz = {0, 0, 0, 0};             // groups 2/3 unused (2D tile)
#if defined(__clang_major__) && __clang_major__ >= 23
    i32x8 gz8 = {0, 0, 0, 0, 0, 0, 0, 0};
    __builtin_amdgcn_tensor_load_to_lds(g0, g1, gz, gz, gz8, 0);
#else
    __builtin_amdgcn_tensor_load_to_lds(g0, g1, gz, gz, 0);
#endif
  }
#endif

  // --- waves stage omega + zero Q padding while the TDM streams A ---
  {
    const float4* o4 = reinterpret_cast<const float4*>(omega);
    for (int i = t; i < (kC * kQ) / 4; i += 256) {
      float4 v = o4[i];
      int base = i * 4, c = base >> 3, q = base & 7;
      sOm[c][q] = v.x; sOm[c][q + 1] = v.y; sOm[c][q + 2] = v.z; sOm[c][q + 3] = v.w;
    }
    for (int i = t; i < kB * 16; i += 256) (&sQm[0][0])[i] = 0.0f;
  }

#if TSVD_USE_TDM
  if ((t >> 5) == 0) __builtin_amdgcn_s_wait_tensorcnt(0);
#else
  // Fallback staging through VGPRs (float4 global loads).
  {
    const float4* g4 = reinterpret_cast<const float4*>(gA);
    for (int i = t; i < kMat / 4; i += 256) {
      float4 v = g4[i];
      int base = i * 4, c = base >> 5, b = base & 31;
      sA[c][b] = v.x; sA[c][b + 1] = v.y; sA[c][b + 2] = v.z; sA[c][b + 3] = v.w;
    }
  }
#endif
  __syncthreads();

  // --- Y = A^T * Omega (32x8, K=256): one output element per thread ---
  {
    int b = t >> 3, q = t & 7;
    float acc = 0.f;
    for (int c = 0; c < kC; ++c) acc = fmaf(sA[c][b], sOm[c][q], acc);
    sYv[b][q] = acc;
  }
  chol_orth32(t, sYv, sQm, sG, sR);          // Q (32x8)

  // --- power iterations (niter = 2) ---
  for (int it = 0; it < 2; ++it) {
    wmma_AxQ(t, sA, sQm, sZ);                // Z = A Q        (WMMA)
    chol_orth256(t, sZ, sG, sR, sGp);        // Qc = orth(Z)
    {                                        // Y = A^T Qc
      int b = t >> 3, q = t & 7;
      float acc = 0.f;
      for (int c = 0; c < kC; ++c) acc = fmaf(sA[c][b], sZ[c][q], acc);
      sYv[b][q] = acc;
    }
    chol_orth32(t, sYv, sQm, sG, sR);        // Q = orth(Y)
  }

  // --- P = A Q; y = P Q^T -> global (in place over A's slot) ---
  wmma_AxQ(t, sA, sQm, sZ);                  // P in sZ        (WMMA)
  __syncthreads();
  wmma_PQt_store(t, sZ, sQm, gA);            // y = P Q^T      (WMMA)
}

extern "C" void kernel_launch(void* const* d_in, const int* in_sizes, int n_in,
                              void* d_out, int out_size, void* d_ws, size_t ws_size,
                              hipStream_t stream) {
  const float* x     = (const float*)d_in[0];
  const float* omega = (const float*)d_in[1];
  float* out  = (float*)d_out;
  float* Abuf = (float*)d_ws;   // needs kM * kMat * 4 = ~102.8 MB scratch

  dim3 blk(256);
  tsvd_transpose_in <<<kC * kH, blk, 0, stream>>>(x, Abuf);
  tsvd_block_kernel <<<kM,      blk, 0, stream>>>(Abuf, omega);
  tsvd_transpose_out<<<kC * kH, blk, 0, stream>>>(Abuf, out);

  (void)in_sizes; (void)n_in; (void)out_size; (void)ws_size;
}